// GCANLoss_13924283973818
// MI455X (gfx1250) — compile-verified
//
#include <hip/hip_runtime.h>
#include <math.h>

typedef __attribute__((ext_vector_type(2))) float v2f;
typedef __attribute__((ext_vector_type(8))) float v8f;

#define B_ 32
#define T_ 32768
#define K_ 8
#define SPLITS 32
#define TCHUNK (T_ / SPLITS)   // 1024
#define NTHREADS 256
#define NWAVES 8

#define LOG2E   1.44269504089f
#define LN2     0.69314718056f

// workspace layout (floats)
#define WS_ACC    0        // B*256 : per-batch 16x16 WMMA accumulators
#define WS_COL    8192     // B*16  : k<8 -> sum nls_neg ; 8+k -> sum g0
#define WS_OVER   8704
#define WS_ASSIGN 8705
#define WS_ORTHO  8706
#define WS_CONTR  8707
#define WS_TOTAL  8708

__global__ void k_zero(float* __restrict__ ws) {
    int i = blockIdx.x * blockDim.x + threadIdx.x;
    if (i < WS_TOTAL) ws[i] = 0.0f;
}

// branch-free jnp.nan_to_num: NaN -> 0, +/-inf -> +/-FLT_MAX
__device__ __forceinline__ float nan0(float x) {
    x = (x != x) ? 0.0f : x;                                   // v_cndmask
    return fminf(fmaxf(x, -3.402823466e38f), 3.402823466e38f); // v_max/v_min
}

// Fused pass over logits/labels: one WMMA per 4 timesteps accumulates BOTH the
// Hungarian-cost contraction (A rows 0..7: PW*nls_pos - nls_neg) and the
// focal-loss contraction (A rows 8..15: g1-g0) against labels (B cols 0..7).
// ml_s is never exactly 1.0 (EPS=0.1), so in the reference pt = 1-probs and
// aw = 0.75 always -> focal element = g1(x) if matched label else g0(x).
// The owning lane evaluates its element ONCE; derived weights are shuffled.
__global__ void __launch_bounds__(NTHREADS) k_phase1(
    const float* __restrict__ logits, const float* __restrict__ labels,
    const float* __restrict__ ovl, const float* __restrict__ ovt,
    float* __restrict__ ws)
{
    const int b    = blockIdx.x / SPLITS;
    const int s    = blockIdx.x % SPLITS;
    const int tid  = threadIdx.x;
    const int lane = tid & 31;
    const int wave = tid >> 5;

    // A layout (16x4 f32): lanes 0-15 -> M=lane, v0=K0,v1=K1 ; lanes 16-31 -> M=lane-16, v0=K2,v1=K3
    // B layout (4x16 f32): lanes 0-15 -> rows 0/1 (v0/v1), lanes 16-31 -> rows 2/3
    const int  m        = lane & 15;             // A row / B col
    const int  kk       = m & 7;                 // logits channel for this row/col
    const bool focal    = (m >= 8);
    const int  halfBase = (lane < 16) ? 0 : 16;  // index into the 32-float (4t x 8k) tile
    const int  src0 = halfBase + kk;             // tile elem for v0 slot
    const int  src1 = halfBase + 8 + kk;         // tile elem for v1 slot

    v8f   acc0 = {};
    v8f   acc1 = {};
    float ccAcc = 0.0f;   // sum nls_neg for own channel (lane&7)
    float cfAcc = 0.0f;   // sum g0      for own channel (lane&7)

    const long baseX = (long)b * T_ * 8 + (long)s * TCHUNK * 8;
    const int  chunksPerWave = TCHUNK / 4 / NWAVES;   // 32

    for (int c = 0; c < chunksPerWave; c += 4) {
        float xv[4], yv[4];
#pragma unroll
        for (int u = 0; u < 4; ++u) {
            long off = baseX + (long)(((c + u) * NWAVES + wave) * 32) + lane;
            xv[u] = logits[off];   // 32 contiguous floats = 4t x 8k tile
            yv[u] = labels[off];
        }
#pragma unroll
        for (int u = 0; u < 4; ++u) {
            // ---- owning lane evaluates its element once ----
            float x    = nan0(xv[u]);
            float ax   = fabsf(x);
            float e    = __builtin_amdgcn_exp2f(ax * -LOG2E);        // exp(-|x|)
            float l    = __builtin_amdgcn_logf(1.0f + e) * LN2;      // log1p(e), e in (0,1]
            float nlsp = l + fmaxf(-x, 0.0f);                        // -log sigmoid(x)
            float nlsn = l + fmaxf(x, 0.0f);                         // -log sigmoid(-x)
            float sab  = __builtin_amdgcn_rcpf(1.0f + e);
            float p    = (x >= 0.0f) ? sab : 1.0f - sab;             // sigmoid(x)
            float base = 0.75f * p * p;
            float aC   = 5.0f * nlsp - nlsn;                         // PW*nlsp - nlsn
            float aF   = base * 0.8f * (nlsp - nlsn);                // g1 - g0
            ccAcc += nlsn;                                           // cost colsum
            cfAcc += base * (0.1f * nlsp + 0.9f * nlsn);             // g0 colsum

            // ---- distribute derived weights + labels into WMMA layouts ----
            float aC0 = __shfl(aC, src0, 32);
            float aF0 = __shfl(aF, src0, 32);
            float aC1 = __shfl(aC, src1, 32);
            float aF1 = __shfl(aF, src1, 32);
            float yb0 = __shfl(yv[u], src0, 32);
            float yb1 = __shfl(yv[u], src1, 32);
            float a0 = focal ? aF0 : aC0;
            float a1 = focal ? aF1 : aC1;
            yb0 = (m < 8) ? yb0 : 0.0f;   // B cols 8..15 zero padding
            yb1 = (m < 8) ? yb1 : 0.0f;

            v2f av = {a0, a1};
            v2f bv = {yb0, yb1};
            if (u & 1)
                acc1 = __builtin_amdgcn_wmma_f32_16x16x4_f32(
                    false, av, false, bv, (short)0, acc1, false, false);
            else
                acc0 = __builtin_amdgcn_wmma_f32_16x16x4_f32(
                    false, av, false, bv, (short)0, acc0, false, false);
        }
    }

    __shared__ float lacc[256];
    __shared__ float lcol[16];
    __shared__ float lov;
    lacc[tid] = 0.0f;
    if (tid < 16) lcol[tid] = 0.0f;
    if (tid == 0) lov = 0.0f;
    __syncthreads();

    // C/D layout: VGPR r, lanes 0-15 -> M=r,N=lane ; lanes 16-31 -> M=r+8,N=lane-16
    const int mrow = (lane < 16) ? 0 : 8;
#pragma unroll
    for (int r = 0; r < 8; ++r)
        atomicAdd(&lacc[(r + mrow) * 16 + m], acc0[r] + acc1[r]);
    atomicAdd(&lcol[lane & 7], ccAcc);
    atomicAdd(&lcol[8 + (lane & 7)], cfAcc);

    // overlap MSE partial for this (b, split): float4 per thread (TCHUNK==1024)
    {
        const long ob4 = ((long)b * T_ + (long)s * TCHUNK) >> 2;
        const float4* ol4 = (const float4*)ovl;
        const float4* ot4 = (const float4*)ovt;
        float4 a = ol4[ob4 + tid];
        float4 t = ot4[ob4 + tid];
        float dx = a.x - t.x, dy = a.y - t.y, dz = a.z - t.z, dw = a.w - t.w;
        atomicAdd(&lov, dx * dx + dy * dy + dz * dz + dw * dw);
    }
    __syncthreads();

    atomicAdd(&ws[WS_ACC + b * 256 + tid], lacc[tid]);
    if (tid < 16) atomicAdd(&ws[WS_COL + b * 16 + tid], lcol[tid]);
    if (tid == 0) atomicAdd(&ws[WS_OVER], lov);
}

// lexicographic permutation decode == itertools.permutations order
__device__ __forceinline__ void decode_perm(int p, int* perm) {
    const int fact[8] = {5040, 720, 120, 24, 6, 2, 1, 1};
    unsigned avail = 0xFFu;
    int rem = p;
#pragma unroll
    for (int k = 0; k < 8; ++k) {
        int d = rem / fact[k]; rem -= d * fact[k];
        unsigned mask = avail;
        for (int j = 0; j < d; ++j) mask &= mask - 1u;
        int col = __ffs(mask) - 1;
        avail &= ~(1u << col);
        perm[k] = col;
    }
}

__global__ void __launch_bounds__(256) k_phase2(float* __restrict__ ws)
{
    __shared__ float costL[64];
    __shared__ float bv[256];
    __shared__ int   bi[256];
    const int b = blockIdx.x;
    const int tid = threadIdx.x;

    if (tid < 64) {
        int k = tid >> 3, n = tid & 7;
        float c = (ws[WS_ACC + b * 256 + k * 16 + n] + ws[WS_COL + b * 16 + k])
                  * (1.0f / (float)T_);
        if (!(fabsf(c) <= 3.0e38f)) c = 100.0f;   // nan_to_num(..., 100.0)
        costL[tid] = c;
    }
    __syncthreads();

    float mv = INFINITY; int mi = 0;
    for (int p = tid; p < 40320; p += 256) {
        int perm[8];
        decode_perm(p, perm);
        float ssum = 0.0f;
#pragma unroll
        for (int k = 0; k < 8; ++k) ssum += costL[k * 8 + perm[k]];
        if (ssum < mv) { mv = ssum; mi = p; }
    }
    bv[tid] = mv; bi[tid] = mi;
    __syncthreads();
    for (int off = 128; off > 0; off >>= 1) {
        if (tid < off) {
            if (bv[tid + off] < bv[tid] ||
                (bv[tid + off] == bv[tid] && bi[tid + off] < bi[tid])) {
                bv[tid] = bv[tid + off]; bi[tid] = bi[tid + off];
            }
        }
        __syncthreads();
    }
    if (tid == 0) {
        int perm[8];
        decode_perm(bi[0], perm);
        float a = 0.0f;
#pragma unroll
        for (int k = 0; k < 8; ++k)
            a += ws[WS_ACC + b * 256 + (8 + k) * 16 + perm[k]]
               + ws[WS_COL + b * 16 + 8 + k];
        atomicAdd(&ws[WS_ASSIGN], a);
    }
}

__global__ void __launch_bounds__(256) k_phase3(const float* __restrict__ attr,
                                                float* __restrict__ ws)
{
    __shared__ float att[2048];
    __shared__ float invn[8];
    __shared__ float ortho_s, contr_s;
    const int b = blockIdx.x;
    const int tid = threadIdx.x;
    for (int i = tid; i < 2048; i += 256) att[i] = attr[b * 2048 + i];
    if (tid == 0) { ortho_s = 0.0f; contr_s = 0.0f; }
    __syncthreads();

    int wave = tid >> 5, lane = tid & 31;
    float ss = 0.0f;
    for (int d = lane; d < 256; d += 32) { float v = att[wave * 256 + d]; ss += v * v; }
    for (int off = 16; off > 0; off >>= 1) ss += __shfl_down(ss, off, 32);
    if (lane == 0) invn[wave] = 1.0f / fmaxf(sqrtf(ss), 1e-12f);
    __syncthreads();

    if (tid < 64) {
        int k = tid >> 3, j = tid & 7;
        float dot = 0.0f;
        for (int d = 0; d < 256; ++d) dot += att[k * 256 + d] * att[j * 256 + d];
        float sim = dot * invn[k] * invn[j];
        float diff = sim - ((k == j) ? 1.0f : 0.0f);
        atomicAdd(&ortho_s, diff * diff);
        if (k != j) atomicAdd(&contr_s, fmaxf(sim + 0.5f, 0.0f));
    }
    __syncthreads();
    if (tid == 0) {
        atomicAdd(&ws[WS_ORTHO], ortho_s);
        atomicAdd(&ws[WS_CONTR], contr_s);
    }
}

__global__ void __launch_bounds__(256) k_final(const float* __restrict__ existence,
                                               const float* __restrict__ ws,
                                               float* __restrict__ out)
{
    __shared__ float red[256];
    const int tid = threadIdx.x;
    float e = existence[tid];            // exactly B*K = 256 elements
    float ax = fabsf(e);
    float ee = __builtin_amdgcn_exp2f(ax * -LOG2E);
    float l  = __builtin_amdgcn_logf(1.0f + ee) * LN2;
    float nlsp = l + fmaxf(-e, 0.0f);
    float nlsn = l + fmaxf(e, 0.0f);
    red[tid] = 4.5f * nlsp + 0.1f * nlsn;   // PW*0.9*nlsp + 0.1*nlsn
    __syncthreads();
    for (int off = 128; off > 0; off >>= 1) {
        if (tid < off) red[tid] += red[tid + off];
        __syncthreads();
    }
    if (tid == 0) {
        float exist_mean = red[0] * (1.0f / 256.0f);
        float assign = ws[WS_ASSIGN] / ((float)B_ * (float)T_ * (float)K_);
        float ortho  = ws[WS_ORTHO]  / ((float)B_ * (float)(K_ * K_));
        float contr  = ws[WS_CONTR]  / ((float)B_ * (float)(K_ * (K_ - 1)));
        float over   = ws[WS_OVER]   / ((float)B_ * (float)T_);
        out[0] = assign + 1.0f * exist_mean + 0.1f * ortho + 0.1f * contr + 0.5f * over;
    }
}

extern "C" void kernel_launch(void* const* d_in, const int* in_sizes, int n_in,
                              void* d_out, int out_size, void* d_ws, size_t ws_size,
                              hipStream_t stream)
{
    const float* logits     = (const float*)d_in[0];
    const float* existence  = (const float*)d_in[1];
    const float* attractors = (const float*)d_in[2];
    const float* gt         = (const float*)d_in[3];
    const float* ovl        = (const float*)d_in[4];
    const float* ovt        = (const float*)d_in[5];
    float* ws  = (float*)d_ws;
    float* out = (float*)d_out;

    k_zero<<<(WS_TOTAL + 255) / 256, 256, 0, stream>>>(ws);
    k_phase1<<<B_ * SPLITS, NTHREADS, 0, stream>>>(logits, gt, ovl, ovt, ws);
    k_phase2<<<B_, 256, 0, stream>>>(ws);
    k_phase3<<<B_, 256, 0, stream>>>(attractors, ws);
    k_final<<<1, 256, 0, stream>>>(existence, ws, out);
}